// Unitrackrion_44220983280269
// MI455X (gfx1250) — compile-verified
//
#include <hip/hip_runtime.h>
#include <math.h>

#define BB 32
#define QQ 4096
#define MM 512
#define TMAXV 256
#define SENTV 257
#define NSEG_PAD 272   // 17 tiles * 16
#define NTILE 17
#define SCALE_F (1.0f / 2202.907169f)   // 1/sqrt(1920^2+1080^2)

// workspace layout (floats)
#define WS_TRK 0
#define WS_SP  32
#define WS_TE  64
#define WS_NU  96
#define WS_NV  128
#define WS_PP  160            // 32 * 272 pred_present flags
// total = 160 + 32*272 = 8864 floats

typedef __attribute__((ext_vector_type(2))) float v2f;
typedef __attribute__((ext_vector_type(8))) float v8f;

__global__ __launch_bounds__(256) void k0_zero(float* __restrict__ ws) {
  for (int i = threadIdx.x; i < WS_PP; i += 256) ws[i] = 0.0f;
}

// ---------------- K1: per-batch segment statistics via WMMA one-hot matmul ----
__global__ __launch_bounds__(256) void k1_stats(const float* __restrict__ pboxes,
                                                const int*   __restrict__ ptid,
                                                const int*   __restrict__ gtid,
                                                float*       __restrict__ ws) {
  const int b   = blockIdx.x;
  const int tid = threadIdx.x;

  __shared__ __align__(16) float2 s_wh[QQ];         // 32 KB, (w,h) interleaved
  __shared__ __align__(8)  int    s_seg[QQ];        // 16 KB
  __shared__ float s_cnt[NSEG_PAD], s_sumw[NSEG_PAD], s_sumh[NSEG_PAD];
  __shared__ float s_mw[NSEG_PAD],  s_mh[NSEG_PAD];
  __shared__ float s_dw[NSEG_PAD],  s_dh[NSEG_PAD];
  __shared__ float s_gt[NSEG_PAD];
  __shared__ float s_acc[8];   // 0:nvalid 1:fn+fp 2:spatial 3:temporal 4:n_unique

  for (int t = tid; t < NSEG_PAD; t += 256) s_gt[t] = 0.0f;
  if (tid < 8) s_acc[tid] = 0.0f;

  const float* pbb = pboxes + (size_t)b * QQ * 4;
  const int*   pib = ptid + b * QQ;
  int nval = 0;
  for (int q = tid; q < QQ; q += 256) {
    float4 bx = ((const float4*)pbb)[q];
    int t = pib[q];
    bool v = t > 0;
    s_seg[q] = v ? t : SENTV;
    s_wh[q]  = make_float2(bx.z - bx.x, bx.w - bx.y);
    nval += v ? 1 : 0;
  }
  __syncthreads();   // s_gt zeroed before atomics; s_seg/s_wh visible
  const int* gib = gtid + b * MM;
  for (int m = tid; m < MM; m += 256) atomicAdd(&s_gt[gib[m]], 1.0f);
  atomicAdd(&s_acc[0], (float)nval);
  __syncthreads();

  const int wave  = tid >> 5;
  const int lane  = tid & 31;
  const int row   = lane & 15;          // A: M (channel); B: N (segment-in-tile)
  const int khalf = (lane >> 4) << 1;   // K base: 0 (lanes 0-15) / 2 (lanes 16-31)
  // branch-free channel selectors (A matrix rows)
  const float is0 = (row == 0) ? 1.0f : 0.0f;
  const float is1 = (row == 1) ? 1.0f : 0.0f;
  const float is2 = (row == 2) ? 1.0f : 0.0f;

  // ---- WMMA pass 1: per-segment [count, sum_w, sum_h] -----------------------
  for (int ti = wave; ti < NTILE; ti += 8) {
    const int t0   = ti * 16;
    const int tseg = t0 + row;          // B one-hot target segment for this lane
    v8f c = {};
    for (int k0 = 0; k0 < QQ; k0 += 4) {
      const int q0 = k0 + khalf;        // even -> 16B aligned float2 pair
      const float4 whp = *(const float4*)(&s_wh[q0]);   // (w0,h0,w1,h1)
      const int2   sgp = *(const int2*)(&s_seg[q0]);    // seg q0, q1
      v2f a, bm;
      a.x  = is0 + is1 * whp.x + is2 * whp.y;
      a.y  = is0 + is1 * whp.z + is2 * whp.w;
      bm.x = (sgp.x == tseg) ? 1.0f : 0.0f;
      bm.y = (sgp.y == tseg) ? 1.0f : 0.0f;
      c = __builtin_amdgcn_wmma_f32_16x16x4_f32(false, a, false, bm,
                                                (short)0, c, false, false);
    }
    if (lane < 16) {
      s_cnt [t0 + lane] = c[0];
      s_sumw[t0 + lane] = c[1];
      s_sumh[t0 + lane] = c[2];
    }
  }
  __syncthreads();

  // ---- means, pred_present, FN -----------------------------------------------
  float fn = 0.0f, nu = 0.0f;
  float* ppb = ws + WS_PP + b * NSEG_PAD;
  for (int t = tid; t < NSEG_PAD; t += 256) {
    const float cnt  = s_cnt[t];
    const float safe = fmaxf(cnt, 1.0f);
    s_mw[t] = s_sumw[t] / safe;
    s_mh[t] = s_sumh[t] / safe;
    const bool real = (t >= 1) && (t <= TMAXV);
    const bool pp   = real && (cnt > 0.5f);
    ppb[t] = pp ? 1.0f : 0.0f;
    if (pp) nu += 1.0f;
    if ((s_gt[t] > 0.5f) && !pp) fn += s_gt[t];
  }
  atomicAdd(&s_acc[4], nu);
  __syncthreads();

  // ---- FP count + transform (w,h) -> (|dev_w|,|dev_h|)*SCALE in place --------
  float fp = 0.0f;
  for (int q = tid; q < QQ; q += 256) {
    const int   sg = s_seg[q];
    const float2 wh = s_wh[q];
    if (sg != SENTV && s_gt[sg] <= 0.5f) fp += 1.0f;
    s_wh[q] = make_float2(fabsf(wh.x - s_mw[sg]) * SCALE_F,
                          fabsf(wh.y - s_mh[sg]) * SCALE_F);
  }
  atomicAdd(&s_acc[1], 0.9f * fn + 0.9f * fp);
  __syncthreads();

  // ---- WMMA pass 2: per-segment [sum|dev_w|, sum|dev_h|] --------------------
  for (int ti = wave; ti < NTILE; ti += 8) {
    const int t0   = ti * 16;
    const int tseg = t0 + row;
    v8f c = {};
    for (int k0 = 0; k0 < QQ; k0 += 4) {
      const int q0 = k0 + khalf;
      const float4 whp = *(const float4*)(&s_wh[q0]);   // (dw0,dh0,dw1,dh1)
      const int2   sgp = *(const int2*)(&s_seg[q0]);
      v2f a, bm;
      a.x  = is0 * whp.x + is1 * whp.y;
      a.y  = is0 * whp.z + is1 * whp.w;
      bm.x = (sgp.x == tseg) ? 1.0f : 0.0f;
      bm.y = (sgp.y == tseg) ? 1.0f : 0.0f;
      c = __builtin_amdgcn_wmma_f32_16x16x4_f32(false, a, false, bm,
                                                (short)0, c, false, false);
    }
    if (lane < 16) { s_dw[t0 + lane] = c[0]; s_dh[t0 + lane] = c[1]; }
  }
  __syncthreads();

  // ---- spatial ---------------------------------------------------------------
  float sp = 0.0f;
  for (int t = tid; t < NSEG_PAD; t += 256) {
    const float cnt = s_cnt[t];
    if ((t >= 1) && (t <= TMAXV) && (cnt > 1.5f))
      sp += (s_dw[t] + s_dh[t]) / fmaxf(cnt, 1.0f);
  }
  atomicAdd(&s_acc[2], sp);

  // ---- temporal: one thread per real track, scan in query order --------------
  {
    const int trk = tid + 1;   // tracks 1..256
    int n = 0;
    float c1x = 0, c1y = 0, c2x = 0, c2y = 0, sum = 0.0f;
    for (int q = 0; q < QQ; ++q) {
      if (s_seg[q] == trk) {
        float4 bx = ((const float4*)pbb)[q];
        const float cx = (bx.x + bx.z) * 0.5f;
        const float cy = (bx.y + bx.w) * 0.5f;
        ++n;
        if (n >= 3) {
          const float ax = (cx - 2.0f * c1x + c2x) * SCALE_F;
          const float ay = (cy - 2.0f * c1y + c2y) * SCALE_F;
          sum += sqrtf(fmaxf(ax * ax + ay * ay, 1e-30f));
        }
        c2x = c1x; c2y = c1y; c1x = cx; c1y = cy;
      }
    }
    if (n > 2) atomicAdd(&s_acc[3], sum / fmaxf((float)n - 2.0f, 1.0f));
  }
  __syncthreads();

  if (tid == 0) {
    const float nvalid = s_acc[0];
    const float nuT    = s_acc[4];
    const float gate   = (nvalid > 1.0f) ? 1.0f : 0.0f;
    const float inv    = 1.0f / fmaxf(nuT, 1.0f);
    atomicAdd(&ws[WS_TRK + b], s_acc[1]);
    ws[WS_SP + b] = gate * s_acc[2] * inv;
    ws[WS_TE + b] = gate * s_acc[3] * inv;
    ws[WS_NU + b] = nuT;
    ws[WS_NV + b] = nvalid;
  }
}

// ---------------- K2: IoU tiles -> localization + switch terms ----------------
__global__ __launch_bounds__(256) void k2_iou(const float* __restrict__ pboxes,
                                              const int*   __restrict__ ptid,
                                              const float* __restrict__ gboxes,
                                              const int*   __restrict__ gtid,
                                              float*       __restrict__ ws) {
  const int b   = blockIdx.y;
  const int tid = threadIdx.x;
  __shared__ float4 s_gb[MM];
  __shared__ float  s_ga[MM];
  __shared__ int    s_gid[MM];
  __shared__ float  s_pp[MM];
  __shared__ float  s_red;
  if (tid == 0) s_red = 0.0f;

  const float* ppb = ws + WS_PP + b * NSEG_PAD;
  for (int m = tid; m < MM; m += 256) {
    float4 g = ((const float4*)(gboxes + (size_t)b * MM * 4))[m];
    s_gb[m] = g;
    s_ga[m] = fmaxf(g.z - g.x, 0.0f) * fmaxf(g.w - g.y, 0.0f);
    const int t = gtid[b * MM + m];
    s_gid[m] = t;
    s_pp[m]  = ppb[t];
  }
  const int q = blockIdx.x * 256 + tid;
  float4 p = ((const float4*)(pboxes + (size_t)b * QQ * 4))[q];
  const int t = ptid[b * QQ + q];
  const bool valid = t > 0;
  const float a1 = fmaxf(p.z - p.x, 0.0f) * fmaxf(p.w - p.y, 0.0f);
  __syncthreads();

  float maxm = -1.0f;
  int   sw   = 0;
  for (int m = 0; m < MM; ++m) {
    const float4 g = s_gb[m];
    const float ltx = fmaxf(p.x, g.x), lty = fmaxf(p.y, g.y);
    const float rbx = fminf(p.z, g.z), rby = fminf(p.w, g.w);
    const float inter = fmaxf(rbx - ltx, 0.0f) * fmaxf(rby - lty, 0.0f);
    float iou = inter / (a1 + s_ga[m] - inter + 1e-6f);
    iou = fminf(fmaxf(iou, 0.0f), 1.0f);
    const bool match = (t == s_gid[m]);
    maxm = match ? fmaxf(maxm, iou) : maxm;
    if (valid && !match && (iou > 0.5f) && (s_pp[m] > 0.5f)) ++sw;
  }
  float part = 1.5f * (float)sw;                  // G_SW * switch count
  if (valid && maxm >= 0.0f) part += 1.0f - maxm; // loc term (match exists <=> gt_present[pid])
  atomicAdd(&s_red, part);
  __syncthreads();
  if (tid == 0) atomicAdd(&ws[WS_TRK + b], s_red);
}

// ---------------- K3: final cross-batch reduction ----------------------------
__global__ void k3_final(const float* __restrict__ ws, float* __restrict__ out) {
  const int b = threadIdx.x;   // 32 lanes = 32 batches, wave32
  const float nv = ws[WS_NV + b];
  const float z  = (nv > 0.0f) ? 1.0f : 0.0f;
  float tr = ws[WS_TRK + b] * z;
  float sp = ws[WS_SP + b] * z;
  float te = ws[WS_TE + b] * z;
  float nu = ws[WS_NU + b] * z;
  for (int o = 16; o >= 1; o >>= 1) {
    tr += __shfl_xor(tr, o, 32);
    sp += __shfl_xor(sp, o, 32);
    te += __shfl_xor(te, o, 32);
    nu += __shfl_xor(nu, o, 32);
  }
  if (b == 0) {
    const float denom = fmaxf(nu, 1.0f);
    float lt  = (nu > 0.0f) ? tr / denom : 0.0f;
    float ls  = (nu > 0.0f) ? sp / denom : 0.0f;
    float lte = (nu > 0.0f) ? te / denom : 0.0f;
    lt  = isfinite(lt)  ? lt  : 0.0f;
    ls  = isfinite(ls)  ? ls  : 0.0f;
    lte = isfinite(lte) ? lte : 0.0f;
    float tot = 2.0f * lt + 1.5f * ls + 1.8f * lte;
    tot = isfinite(tot) ? tot : 0.0f;
    out[0] = tot; out[1] = lt; out[2] = ls; out[3] = lte;
  }
}

extern "C" void kernel_launch(void* const* d_in, const int* in_sizes, int n_in,
                              void* d_out, int out_size, void* d_ws, size_t ws_size,
                              hipStream_t stream) {
  (void)in_sizes; (void)n_in; (void)out_size; (void)ws_size;
  const float* pboxes = (const float*)d_in[0];
  // d_in[1] = pred_logits: unused by the reference loss
  const int*   ptid   = (const int*)d_in[2];
  const float* gboxes = (const float*)d_in[3];
  const int*   gtid   = (const int*)d_in[4];
  float* ws  = (float*)d_ws;
  float* out = (float*)d_out;

  k0_zero<<<1, 256, 0, stream>>>(ws);
  k1_stats<<<BB, 256, 0, stream>>>(pboxes, ptid, gtid, ws);
  dim3 g2(QQ / 256, BB);
  k2_iou<<<g2, 256, 0, stream>>>(pboxes, ptid, gboxes, gtid, ws);
  k3_final<<<1, 32, 0, stream>>>(ws, out);
}